// DeformableConv2d_88519275971164
// MI455X (gfx1250) — compile-verified
//
#include <hip/hip_runtime.h>
#include <math.h>

// ---------------------------------------------------------------------------
// Fully fused Deformable Conv2d for MI455X (gfx1250, wave32, WMMA).
//
// Single kernel, per block of 16 spatial positions (one b,h row segment):
//   Phase 0: regular im2col tile  cols[576][16] in LDS         (fixed taps)
//   Phase 1: offset GEMM  off[18][16] = PGW(18x576) x cols     (WMMA, waves 0-1)
//   Phase 2: bilinear gather using LDS offsets -> overwrite cols with
//            deformed im2col tile                              (VALU gather)
//   Phase 3: main GEMM  out[64][16] = W(64x576) x cols + bias  (WMMA, 4 waves)
//
// GEMM shape overall: M=64 (+18), N=B*H*W=131072, K=576, fp32 via
// V_WMMA_F32_16X16X4_F32 (matches reference fp32 numerics).
// ---------------------------------------------------------------------------

typedef __attribute__((ext_vector_type(2))) float v2f;
typedef __attribute__((ext_vector_type(8))) float v8f;

#define B_    8
#define C_    64
#define O_    64
#define H_    128
#define W_    128
#define HW_   (H_ * W_)
#define K_    9
#define CK_   576          // C_ * K_
#define OFFC_ 18           // 2 * K_

#define NPOS  16           // spatial positions per block
#define CSTR  17           // cols row stride (pad -> conflict-free B-frag reads)

__global__ __launch_bounds__(128) void fused_deform_wmma_kernel(
    const float* __restrict__ x,
    const float* __restrict__ pgw,   // [18][CK_]  == (18,64,3,3) flat
    const float* __restrict__ pgb,   // [18]
    const float* __restrict__ wgt,   // [64][CK_]  == (64,64,3,3) flat
    const float* __restrict__ bias,  // [64]
    float* __restrict__ out)
{
  __shared__ float cols[CK_ * CSTR];      // 576*17*4 = 39168 B (reused twice)
  __shared__ float sOff[OFFC_][NPOS];     // offset-conv result, 1152 B

  const int tid = threadIdx.x;
  const int s0  = blockIdx.x * NPOS;      // first linear position
  const int b   = s0 >> 14;
  const int hw0 = s0 & (HW_ - 1);
  const int h   = hw0 >> 7;
  const int w0  = hw0 & (W_ - 1);
  const float* xb = x + (size_t)b * C_ * HW_;

  const int wave = tid >> 5;              // 0..3
  const int lane = tid & 31;
  const int mn   = lane & 15;             // M for A-frags, N for B/C/D-frags
  const int half = lane >> 4;

  // ---- Phase 0: regular (fixed-tap) im2col tile into LDS ------------------
  // p -> (row = c*9+k, n); 16 lanes cover 16 contiguous w -> coalesced loads.
  for (int p = tid; p < CK_ * NPOS; p += 128) {
    const int row = p >> 4;               // c*9 + k
    const int n   = p & 15;
    const int c   = row / 9;
    const int k   = row - 9 * c;
    const int kh  = k / 3, kw = k % 3;
    const int hy  = h + kh - 1;
    const int wx  = w0 + n + kw - 1;
    float v = 0.0f;
    if (hy >= 0 && hy < H_ && wx >= 0 && wx < W_)
      v = xb[c * HW_ + hy * W_ + wx];
    cols[row * CSTR + n] = v;
  }
  __syncthreads();

  // ---- Phase 1: offset conv as WMMA GEMM (waves 0-1; wave-uniform branch) -
  // M tile = 32 rows, only 18 valid: clamp A-row pointer (no OOB reads);
  // invalid D rows are never stored (D[m,*] depends only on A[m,*]).
  if (wave < 2) {
    const int orow = wave * 16 + mn;                       // 0..31
    const int crow = orow < OFFC_ ? orow : (OFFC_ - 1);
    const float* prow = pgw + (size_t)crow * CK_;

    v8f acc0 = {};
    v8f acc1 = {};
    for (int kb = 0; kb < CK_; kb += 8) {
      {
        const int ka = kb + 2 * half;
        v2f a;  a.x  = prow[ka];               a.y  = prow[ka + 1];
        v2f bm; bm.x = cols[ka * CSTR + mn];   bm.y = cols[(ka + 1) * CSTR + mn];
        acc0 = __builtin_amdgcn_wmma_f32_16x16x4_f32(
            false, a, false, bm, (short)0, acc0, false, false);
      }
      {
        const int ka = kb + 4 + 2 * half;
        v2f a;  a.x  = prow[ka];               a.y  = prow[ka + 1];
        v2f bm; bm.x = cols[ka * CSTR + mn];   bm.y = cols[(ka + 1) * CSTR + mn];
        acc1 = __builtin_amdgcn_wmma_f32_16x16x4_f32(
            false, a, false, bm, (short)0, acc1, false, false);
      }
    }
    #pragma unroll
    for (int r = 0; r < 8; ++r) {
      const int ch = wave * 16 + r + 8 * half;             // D row -> channel
      if (ch < OFFC_)
        sOff[ch][mn] = acc0[r] + acc1[r] + pgb[ch];
    }
  }
  __syncthreads();   // offsets ready; all waves done reading phase-0 cols

  // ---- Phase 2: bilinear gather -> overwrite cols with deformed tile ------
  for (int p = tid; p < K_ * NPOS; p += 128) {
    const int k  = p / NPOS;
    const int n  = p % NPOS;
    const int w  = w0 + n;
    const int kh = k / 3, kw = k % 3;
    const float dy = sOff[2 * k][n];
    const float dx = sOff[2 * k + 1][n];
    const float py = dy + (float)(h - 1 + kh);   // base_y = h - PAD, ky = kh
    const float px = dx + (float)(w - 1 + kw);
    const float y0 = floorf(py);
    const float x0 = floorf(px);

    float bw[4];
    int   bi[4];
    #pragma unroll
    for (int t = 0; t < 4; ++t) {
      const float yc = y0 + (float)(t >> 1);
      const float xc = x0 + (float)(t & 1);
      const float wy = 1.0f - fabsf(py - yc);
      const float wx = 1.0f - fabsf(px - xc);
      const bool valid = (yc >= 0.0f) && (yc < (float)H_) &&
                         (xc >= 0.0f) && (xc < (float)W_);
      const int yi = (int)fminf(fmaxf(yc, 0.0f), (float)(H_ - 1));
      const int xi = (int)fminf(fmaxf(xc, 0.0f), (float)(W_ - 1));
      bw[t] = valid ? (wy * wx) : 0.0f;
      bi[t] = yi * W_ + xi;
    }

    for (int c = 0; c < C_; ++c) {
      const float* xc = xb + c * HW_;
      const float g = bw[0] * xc[bi[0]] + bw[1] * xc[bi[1]] +
                      bw[2] * xc[bi[2]] + bw[3] * xc[bi[3]];
      cols[(c * K_ + k) * CSTR + n] = g;
    }
  }
  __syncthreads();

  // ---- Phase 3: main WMMA GEMM, one 16x16 f32 tile per wave ---------------
  // A 16x4 f32: lanes 0-15 -> M; VGPR0 = K+0 / K+2 (half), VGPR1 = K+1 / K+3.
  // B 4x16 f32: lanes 0-15 -> N; halves hold rows K / K+2, K+1 / K+3.
  const int obase = wave * 16;
  const float* wrow = wgt + (size_t)(obase + mn) * CK_;

  v8f acc0 = {};
  v8f acc1 = {};
  for (int kb = 0; kb < CK_; kb += 8) {
    {
      const int ka = kb + 2 * half;
      v2f a;  a.x  = wrow[ka];               a.y  = wrow[ka + 1];   // b64 load
      v2f bm; bm.x = cols[ka * CSTR + mn];   bm.y = cols[(ka + 1) * CSTR + mn];
      acc0 = __builtin_amdgcn_wmma_f32_16x16x4_f32(
          false, a, false, bm, (short)0, acc0, false, false);
    }
    {
      const int ka = kb + 4 + 2 * half;
      v2f a;  a.x  = wrow[ka];               a.y  = wrow[ka + 1];
      v2f bm; bm.x = cols[ka * CSTR + mn];   bm.y = cols[(ka + 1) * CSTR + mn];
      acc1 = __builtin_amdgcn_wmma_f32_16x16x4_f32(
          false, a, false, bm, (short)0, acc1, false, false);
    }
  }

  // ---- Store: C/D layout VGPR r -> M = r + 8*half, lane&15 -> N -----------
  #pragma unroll
  for (int r = 0; r < 8; ++r) {
    const int o = obase + r + 8 * half;
    const float v = acc0[r] + acc1[r] + bias[o];
    out[((size_t)(b * O_ + o) << 14) + hw0 + mn] = v;
  }
}

// ---------------------------------------------------------------------------
extern "C" void kernel_launch(void* const* d_in, const int* in_sizes, int n_in,
                              void* d_out, int out_size, void* d_ws, size_t ws_size,
                              hipStream_t stream) {
  const float* x    = (const float*)d_in[0];
  const float* pgw  = (const float*)d_in[1];
  const float* pgb  = (const float*)d_in[2];
  const float* wgt  = (const float*)d_in[3];
  const float* bias = (const float*)d_in[4];
  float* out = (float*)d_out;

  fused_deform_wmma_kernel<<<(B_ * HW_) / NPOS, 128, 0, stream>>>(
      x, pgw, pgb, wgt, bias, out);
}